// AUFusion_49014166782574
// MI455X (gfx1250) — compile-verified
//
#include <hip/hip_runtime.h>

// ---------------------------------------------------------------------------
// AUFusion for MI455X (gfx1250, wave32).
//
// Roofline: 369 MB streamed @ 23.3 TB/s ~= 16 us; 151 MFLOP is negligible.
// Kernel 1 (bandwidth-bound): one wave32 per sample, float4 (b128) coalesced
//   loads of eyebrow/mouth/gcn rows, per-lane FMA, wave butterfly reduction,
//   LayerNorm; writes normed features padded to stride 12 into d_ws.
// Kernel 2 (CDNA5 matrix path): final Linear(9->5) as a real shared-weight
//   GEMM via V_WMMA_F32_16X16X4_F32, 16 samples per wave, K=12 (padded) in
//   3 WMMA steps, bias preloaded into the C accumulator.
// ---------------------------------------------------------------------------

#define K_DIM   512
#define KV4     (K_DIM / 4)   // 128 float4 per K-row
#define NF      9             // IN_FEATURES
#define NC      5             // NUM_CLASSES
#define NF_PAD  12            // padded feature stride for WMMA (3 x K=4)
#define LN_EPS  1e-6f

typedef __attribute__((ext_vector_type(2))) float v2f;
typedef __attribute__((ext_vector_type(8))) float v8f;

__device__ __forceinline__ float wave32_reduce_add(float v) {
#pragma unroll
    for (int off = 16; off > 0; off >>= 1)
        v += __shfl_xor(v, off, 32);   // wave32: width MUST be 32 on gfx1250
    return v;
}

// One wave32 per sample. FUSED=true: write final logits directly (VALU
// epilogue). FUSED=false: write LayerNorm output, zero-padded to stride 12,
// into workspace for the WMMA linear kernel.
template <bool FUSED>
__global__ __launch_bounds__(256) void aufusion_features_kernel(
    const float* __restrict__ eyebrow,   // [B, 512]
    const float* __restrict__ mouth,     // [B, 512]
    const float* __restrict__ gcn,       // [B, 9, 512]
    const float* __restrict__ ln_w,      // [9]
    const float* __restrict__ ln_b,      // [9]
    const float* __restrict__ lin_w,     // [5, 9]
    const float* __restrict__ lin_b,     // [5]
    float* __restrict__ dst,             // FUSED ? out[B,5] : normed[B,12]
    int B)
{
    const int wave = (int)((blockIdx.x * blockDim.x + threadIdx.x) >> 5);
    const int lane = (int)(threadIdx.x & 31);
    if (wave >= B) return;               // wave-uniform branch
    const int b = wave;

    const float4* e4 = (const float4*)eyebrow + (size_t)b * KV4;
    const float4* m4 = (const float4*)mouth   + (size_t)b * KV4;
    const float4* g4 = (const float4*)gcn     + (size_t)b * NF * KV4;

    // Signal rows are reused across gcn channels: keep them in registers.
    float4 ev[4], mv[4];
#pragma unroll
    for (int i = 0; i < 4; ++i) ev[i] = e4[i * 32 + lane];
#pragma unroll
    for (int i = 0; i < 4; ++i) mv[i] = m4[i * 32 + lane];

    float acc[NF];
#pragma unroll
    for (int c = 0; c < NF; ++c) acc[c] = 0.0f;

#pragma unroll
    for (int c = 0; c < NF; ++c) {
#pragma unroll
        for (int i = 0; i < 4; ++i) {
            const float4 g = g4[(size_t)c * KV4 + i * 32 + lane];
            const float4 s = (c < 3) ? ev[i] : mv[i];
            acc[c] += g.x * s.x + g.y * s.y + g.z * s.z + g.w * s.w;
        }
    }

    // Butterfly-reduce each channel across the wave: all lanes get the sums.
    float f[NF];
#pragma unroll
    for (int c = 0; c < NF; ++c) f[c] = wave32_reduce_add(acc[c]);

    // LayerNorm over 9 features (computed redundantly on every lane).
    float mu = 0.0f;
#pragma unroll
    for (int c = 0; c < NF; ++c) mu += f[c];
    mu *= (1.0f / (float)NF);
    float var = 0.0f;
#pragma unroll
    for (int c = 0; c < NF; ++c) { const float d = f[c] - mu; var += d * d; }
    var *= (1.0f / (float)NF);
    const float inv = rsqrtf(var + LN_EPS);

    float n[NF];
#pragma unroll
    for (int c = 0; c < NF; ++c)
        n[c] = (f[c] - mu) * inv * ln_w[c] + ln_b[c];

    if (FUSED) {
        // Lanes 0..4 each produce one class logit.
        if (lane < NC) {
            float o = lin_b[lane];
#pragma unroll
            for (int c = 0; c < NF; ++c) o += n[c] * lin_w[lane * NF + c];
            dst[(size_t)b * NC + lane] = o;
        }
    } else {
        // Lane 0 writes the padded (stride-12) normed row: three b128 stores.
        if (lane == 0) {
            float4* o = (float4*)(dst + (size_t)b * NF_PAD);
            o[0] = make_float4(n[0], n[1], n[2], n[3]);
            o[1] = make_float4(n[4], n[5], n[6], n[7]);
            o[2] = make_float4(n[8], 0.0f, 0.0f, 0.0f);
        }
    }
}

// Final Linear(9->5) via V_WMMA_F32_16X16X4_F32. One wave32 per 16 samples.
// A tile: 16 samples x K=4 of normed features (padded stride 12, so all
// loads are in-bounds aligned b64). B tile: lin_weight^T chunk (shared),
// zero outside the valid 9x5 region via clamped addresses + selects (no EXEC
// divergence before the WMMA — EXEC must be all-ones). C preloaded with bias.
__global__ __launch_bounds__(256) void aufusion_linear_wmma_kernel(
    const float* __restrict__ normed,   // [B, 12] padded
    const float* __restrict__ lin_w,    // [5, 9]
    const float* __restrict__ lin_b,    // [5]
    float* __restrict__ out,            // [B, 5]
    int B)
{
    const int wave = (int)((blockIdx.x * blockDim.x + threadIdx.x) >> 5);
    const int lane = (int)(threadIdx.x & 31);
    const int s0 = wave * 16;
    if (s0 >= B) return;                 // wave-uniform branch
    const int m    = lane & 15;          // A row (sample) / B,C col (class)
    const int half = lane >> 4;          // K sub-pair selector

    // Bias into accumulator columns 0..4 (clamped address + select).
    const float bias_ld = lin_b[(m < NC) ? m : 0];
    const float bias    = (m < NC) ? bias_ld : 0.0f;
    v8f c;
#pragma unroll
    for (int r = 0; r < 8; ++r) c[r] = bias;

#pragma unroll
    for (int s = 0; s < 3; ++s) {
        const int k = 4 * s + 2 * half;  // this lane's K pair within [0,12)

        // A[m][k], A[m][k+1]: padded rows -> unconditional aligned b64 load.
        const float2 arow = *(const float2*)(normed + (size_t)(s0 + m) * NF_PAD + k);
        v2f a; a.x = arow.x; a.y = arow.y;

        // B[k][n] = lin_w[n][k]; zero outside the 5x9 valid region.
        const bool v0 = (m < NC) && (k     < NF);
        const bool v1 = (m < NC) && (k + 1 < NF);
        const float w0 = lin_w[v0 ? (m * NF + k)     : 0];
        const float w1 = lin_w[v1 ? (m * NF + k + 1) : 0];
        v2f bb; bb.x = v0 ? w0 : 0.0f; bb.y = v1 ? w1 : 0.0f;

        // 8 args: (neg_a, A, neg_b, B, c_mod, C, reuse_a, reuse_b)
        c = __builtin_amdgcn_wmma_f32_16x16x4_f32(
            false, a, false, bb, (short)0, c, false, false);
    }

    // D layout: VGPR r on lane L -> row (r + 8*half), col (L & 15).
    if (m < NC) {
#pragma unroll
        for (int r = 0; r < 8; ++r) {
            const int row = r + 8 * half;
            out[(size_t)(s0 + row) * NC + m] = c[r];
        }
    }
}

extern "C" void kernel_launch(void* const* d_in, const int* in_sizes, int n_in,
                              void* d_out, int out_size, void* d_ws, size_t ws_size,
                              hipStream_t stream) {
    const float* eyebrow = (const float*)d_in[0];
    const float* mouth   = (const float*)d_in[1];
    const float* gcn     = (const float*)d_in[2];
    const float* ln_w    = (const float*)d_in[3];
    const float* ln_b    = (const float*)d_in[4];
    const float* lin_w   = (const float*)d_in[5];
    const float* lin_b   = (const float*)d_in[6];
    float* out = (float*)d_out;

    const int B = in_sizes[0] / K_DIM;

    const dim3 blk(256);                          // 8 wave32s per block
    const int waves_per_block = 256 / 32;
    const dim3 grd_feat((unsigned)((B + waves_per_block - 1) / waves_per_block));

    const size_t ws_need = (size_t)B * NF_PAD * sizeof(float);
    if (ws_size >= ws_need && (B % 16) == 0) {
        float* normed = (float*)d_ws;
        aufusion_features_kernel<false><<<grd_feat, blk, 0, stream>>>(
            eyebrow, mouth, gcn, ln_w, ln_b, lin_w, lin_b, normed, B);
        const int samples_per_block = 16 * waves_per_block;   // 128
        const dim3 grd_lin((unsigned)((B + samples_per_block - 1) / samples_per_block));
        aufusion_linear_wmma_kernel<<<grd_lin, blk, 0, stream>>>(
            normed, lin_w, lin_b, out, B);
    } else {
        aufusion_features_kernel<true><<<grd_feat, blk, 0, stream>>>(
            eyebrow, mouth, gcn, ln_w, ln_b, lin_w, lin_b, out, B);
    }
}